// SLL_67989332296064
// MI455X (gfx1250) — compile-verified
//
#include <hip/hip_runtime.h>
#include <stdint.h>

// ---------------------------------------------------------------------------
// SLL cost: ragged CSR gather + per-net OR-mask + table lookup + weighted sum.
// Memory-bound (~45 MB HBM/call; pos gather L2-resident on 192MB L2).
// 1 thread/net; block stages its contiguous pin slab into LDS with gfx1250
// async-to-LDS copies (global_load_async_to_lds_b32, ASYNCcnt), overlapped
// with per-net metadata loads; deterministic two-pass reduction.
// ---------------------------------------------------------------------------

#define BLOCK   256        // 8 waves (wave32)
#define PIN_CAP 8192       // 32 KB LDS pin slab (avg slab = 1024 pins)

#if defined(__HIP_DEVICE_COMPILE__) && __has_builtin(__builtin_amdgcn_global_load_async_to_lds_b32)
#define HAVE_ASYNC_LDS 1
typedef __attribute__((address_space(1))) int global_int;
typedef __attribute__((address_space(3))) int lds_int;
#else
#define HAVE_ASYNC_LDS 0
#endif

__device__ __forceinline__ int bin_clamp(float v, float lo_f, float div, int nmax) {
    // floor((v - lo)/div) clipped to [0, nmax] — matches reference exactly
    float b = floorf((v - lo_f) / div);
    b = fminf(fmaxf(b, 0.0f), (float)nmax);
    return (int)b;
}

__device__ __forceinline__ void wait_async_lds() {
#if __has_builtin(__builtin_amdgcn_s_wait_asynccnt)
    __builtin_amdgcn_s_wait_asynccnt(0);
#else
    asm volatile("s_wait_asynccnt 0x0" ::: "memory");
#endif
}

__global__ __launch_bounds__(BLOCK)
void sll_per_net_kernel(const float* __restrict__ pos,
                        const int*   __restrict__ flat_netpin,
                        const int*   __restrict__ netpin_start,
                        const float* __restrict__ net_weights,
                        const unsigned char* __restrict__ net_mask,
                        const int*   __restrict__ sll_table,
                        const float* __restrict__ xl_p,
                        const float* __restrict__ yl_p,
                        const float* __restrict__ sw_p,
                        const float* __restrict__ sh_p,
                        const int*   __restrict__ nsx_p,
                        const int*   __restrict__ nsy_p,
                        float* __restrict__ block_partials,
                        int num_nets, int n_nodes)
{
    __shared__ int   s_table[16];
    __shared__ __align__(16) int s_pins[PIN_CAP];
    __shared__ float s_red[BLOCK / 32];
    __shared__ int   s_lo, s_cnt;

    const int tid  = (int)threadIdx.x;
    const int net0 = (int)blockIdx.x * BLOCK;
    const int netE = min(net0 + BLOCK, num_nets);

    // uniform scalar params (single-element device arrays)
    const float xl  = xl_p[0];
    const float yl  = yl_p[0];
    const float sw  = sw_p[0];
    const float sh  = sh_p[0];
    const int   nsx = nsx_p[0];
    const int   nsy = nsy_p[0];

    // prefetch this block's weight/mask lines early (global_prefetch_b8)
    if (net0 + tid < num_nets) {
        __builtin_prefetch(net_weights + net0 + tid, 0, 0);
        __builtin_prefetch(net_mask    + net0 + tid, 0, 0);
    }

    if (tid < 16) s_table[tid] = sll_table[tid];
    if (tid == 0) {
        int lo = netpin_start[net0];
        s_lo  = lo;
        s_cnt = netpin_start[netE] - lo;
    }
    __syncthreads();

    const int  lo     = s_lo;
    const int  cnt    = s_cnt;
    const bool staged = (cnt <= PIN_CAP);

    // 1) kick off the async slab copy (memory -> LDS, no VGPR round trip,
    //    tracked by ASYNCcnt) ...
    if (staged) {
#if HAVE_ASYNC_LDS
        for (int i = tid; i < cnt; i += BLOCK) {
            __builtin_amdgcn_global_load_async_to_lds_b32(
                (global_int*)(flat_netpin + lo + i),
                (lds_int*)(&s_pins[i]),
                /*imm offset*/0, /*cpol*/0);
        }
#else
        for (int i = tid; i < cnt; i += BLOCK)
            s_pins[i] = flat_netpin[lo + i];
#endif
    }

    // 2) ... overlap it with this thread's independent per-net metadata loads
    const int net = net0 + tid;
    int   s = 0, e = 0;
    float wm = 0.0f;
    if (net < num_nets) {
        s  = netpin_start[net];
        e  = netpin_start[net + 1];
        wm = net_weights[net] * (net_mask[net] ? 1.0f : 0.0f);
    }

    // 3) now require the slab
    if (staged) {
#if HAVE_ASYNC_LDS
        wait_async_lds();       // this wave's async copies landed in LDS
#endif
        __syncthreads();        // all waves' copies visible
    }

    float local = 0.0f;
    if (net < num_nets) {
        unsigned mask = 0u;
        if (nsx > 1) {
            // generic 2-D SLR grid: need both x and y
            for (int p = s; p < e; ++p) {
                const int idx = staged ? s_pins[p - lo] : flat_netpin[p];
                const float x = pos[idx];
                const float y = pos[idx + n_nodes];
                const int bx = bin_clamp(x, xl, sw, nsx - 1);
                const int by = bin_clamp(y, yl, sh, nsy - 1);
                mask |= 1u << (bx * nsy + by);
            }
        } else {
            // num_slrX==1 fast path: binx==0 always, gather only y
            for (int p = s; p < e; ++p) {
                const int idx = staged ? s_pins[p - lo] : flat_netpin[p];
                const float y = pos[idx + n_nodes];
                const int by = bin_clamp(y, yl, sh, nsy - 1);
                mask |= 1u << by;
            }
        }
        local = (float)s_table[mask & 15u] * wm;
    }

    // wave32 shuffle reduction (no barriers), then 8 wave-sums via LDS
    #pragma unroll
    for (int off = 16; off > 0; off >>= 1)
        local += __shfl_down(local, off, 32);
    if ((tid & 31) == 0) s_red[tid >> 5] = local;
    __syncthreads();
    if (tid == 0) {
        float t = 0.0f;
        #pragma unroll
        for (int j = 0; j < BLOCK / 32; ++j) t += s_red[j];   // fixed order
        block_partials[blockIdx.x] = t;
    }
}

__global__ __launch_bounds__(256)
void sll_reduce_kernel(const float* __restrict__ partials, int n,
                       float* __restrict__ out)
{
    __shared__ float s_red[256];
    const int tid = (int)threadIdx.x;
    float v = 0.0f;
    for (int i = tid; i < n; i += 256) v += partials[i];
    s_red[tid] = v;
    __syncthreads();
    #pragma unroll
    for (int off = 128; off > 0; off >>= 1) {
        if (tid < off) s_red[tid] += s_red[tid + off];
        __syncthreads();
    }
    if (tid == 0) out[0] = s_red[0];
}

extern "C" void kernel_launch(void* const* d_in, const int* in_sizes, int n_in,
                              void* d_out, int out_size, void* d_ws, size_t ws_size,
                              hipStream_t stream)
{
    const float*         pos          = (const float*)d_in[0];
    const int*           flat_netpin  = (const int*)d_in[1];
    const int*           netpin_start = (const int*)d_in[2];
    const float*         net_weights  = (const float*)d_in[3];
    const unsigned char* net_mask     = (const unsigned char*)d_in[4];
    const int*           sll_table    = (const int*)d_in[5];
    const float*         xl_p         = (const float*)d_in[6];
    const float*         yl_p         = (const float*)d_in[7];
    const float*         sw_p         = (const float*)d_in[8];
    const float*         sh_p         = (const float*)d_in[9];
    const int*           nsx_p        = (const int*)d_in[10];
    const int*           nsy_p        = (const int*)d_in[11];

    const int num_nets = in_sizes[2] - 1;   // netpin_start has num_nets+1 entries
    const int n_nodes  = in_sizes[0] / 2;   // pos = [x(0:N), y(N:2N)]

    const int nblocks = (num_nets + BLOCK - 1) / BLOCK;
    float* partials = (float*)d_ws;         // nblocks floats (~23 KB) << ws_size

    sll_per_net_kernel<<<nblocks, BLOCK, 0, stream>>>(
        pos, flat_netpin, netpin_start, net_weights, net_mask, sll_table,
        xl_p, yl_p, sw_p, sh_p, nsx_p, nsy_p,
        partials, num_nets, n_nodes);

    sll_reduce_kernel<<<1, 256, 0, stream>>>(partials, nblocks, (float*)d_out);
}